// rank_79061757985026
// MI455X (gfx1250) — compile-verified
//
#include <hip/hip_runtime.h>
#include <stdint.h>

// ---------------------------------------------------------------------------
// Per-row (512 contiguous fp32) select-256th-largest + scale kernel, MI455X.
//  - one wave32 per row, 16 elems/lane
//  - row staged to LDS with gfx1250 async global->LDS B128 copies (ASYNCcnt),
//    double buffered so next row streams while current row is radix-selected
//  - exact kth-largest via 32-step bitwise radix select; per-step count is
//    ballot_w32 (1 VALU cmp/elem) + scalar popcount on the SALU pipe
//  - streaming data: NT hint on loads, NT stores (keep 192MB L2 clean)
// ---------------------------------------------------------------------------

#define ROW_LEN 512
#define K_SEL 256u
#define WAVES_PER_BLOCK 8
#define BLOCK_THREADS 256
#define LANE_F4 4  // 4 x float4 = 16 floats per lane

typedef int b128_t __attribute__((vector_size(16)));      // builtin proto type
typedef float f32x4 __attribute__((ext_vector_type(4)));  // native fp vec4

__device__ __forceinline__ uint32_t ballot32(bool p) {
#if __has_builtin(__builtin_amdgcn_ballot_w32)
  return __builtin_amdgcn_ballot_w32(p);
#else
  return (uint32_t)__ballot(p);
#endif
}

template <int N>
__device__ __forceinline__ void wait_asynccnt() {
#if __has_builtin(__builtin_amdgcn_s_wait_asynccnt)
  __builtin_amdgcn_s_wait_asynccnt(N);
#else
  asm volatile("s_wait_asynccnt %0" ::"i"(N) : "memory");
#endif
  asm volatile("" ::: "memory");  // keep LDS reads below the wait
}

// 16B async copy global -> LDS (per lane), TH=NT (read-once streaming data).
__device__ __forceinline__ void async_copy_b128(const float* g, float* lds) {
#if __has_builtin(__builtin_amdgcn_global_load_async_to_lds_b128)
  b128_t* gg = (b128_t*)g;  // generic reinterpret (drops const)
  b128_t* ll = (b128_t*)lds;
  __builtin_amdgcn_global_load_async_to_lds_b128(
      (__attribute__((address_space(1))) b128_t*)gg,
      (__attribute__((address_space(3))) b128_t*)ll, 0, /*cpol TH_NT*/ 1);
#else
  *(f32x4*)lds = *(const f32x4*)g;  // correct, loses overlap
#endif
}

__device__ __forceinline__ void store_nt_f4(float* p, f32x4 v) {
#if __has_builtin(__builtin_nontemporal_store)
  __builtin_nontemporal_store(v, (f32x4*)p);
#else
  *(f32x4*)p = v;
#endif
}

__global__ void __launch_bounds__(BLOCK_THREADS)
topk_scale_kernel(const float* __restrict__ in, float* __restrict__ out,
                  int nrows) {
  __shared__ __align__(16) float smem[WAVES_PER_BLOCK][2][ROW_LEN];

  const int lane = threadIdx.x & 31;
  const int wave = threadIdx.x >> 5;
  float* const buf0 = &smem[wave][0][0];
  float* const buf1 = &smem[wave][1][0];

  const int rowStride = gridDim.x * WAVES_PER_BLOCK;
  int row = blockIdx.x * WAVES_PER_BLOCK + wave;
  if (row >= nrows) return;  // wave-uniform: EXEC stays all-ones per wave

  // Prologue: stage first row into buffer 0.
  {
    const float* g = in + (size_t)row * ROW_LEN;
#pragma unroll
    for (int c = 0; c < LANE_F4; ++c) {
      const int f4 = c * 32 + lane;  // coalesced 512B per instruction
      async_copy_b128(g + f4 * 4, buf0 + f4 * 4);
    }
  }

  int parity = 0;
  for (; row < nrows; row += rowStride) {
    const int nxt = row + rowStride;
    float* const cur = parity ? buf1 : buf0;
    float* const nbuf = parity ? buf0 : buf1;

    if (nxt < nrows) {
      // Prefetch next row into the other buffer, then wait only for the
      // current row's 4 transfers (in-order completion on ASYNCcnt).
      const float* g = in + (size_t)nxt * ROW_LEN;
#pragma unroll
      for (int c = 0; c < LANE_F4; ++c) {
        const int f4 = c * 32 + lane;
        async_copy_b128(g + f4 * 4, nbuf + f4 * 4);
      }
      wait_asynccnt<4>();
    } else {
      wait_asynccnt<0>();
    }

    // Pull this wave's row from LDS; build order-preserving uint keys.
    uint32_t key[16];
#pragma unroll
    for (int c = 0; c < LANE_F4; ++c) {
      const int f4 = c * 32 + lane;
      const f32x4 v = *(const f32x4*)(cur + f4 * 4);  // ds_load_b128
#pragma unroll
      for (int j = 0; j < 4; ++j) {
        const uint32_t b = __float_as_uint(v[j]);
        key[c * 4 + j] = b ^ ((uint32_t)((int32_t)b >> 31) | 0x80000000u);
      }
    }

    // Bitwise radix select of the K-th largest key:
    //   t <- t|bit  iff  count(key >= t|bit) >= K   (monotone, no rank state)
    uint32_t t = 0u;
#pragma unroll 1
    for (int bit = 31; bit >= 0; --bit) {
      const uint32_t cand = t | (1u << bit);
      uint32_t cnt = 0u;
#pragma unroll
      for (int i = 0; i < 16; ++i)
        cnt += __popc(ballot32(key[i] >= cand));  // v_cmp + s_bcnt/s_add
      if (cnt >= K_SEL) t = cand;
    }
    const uint32_t tb = (t & 0x80000000u) ? (t ^ 0x80000000u) : ~t;
    const float thresh = __uint_as_float(tb);  // bit-exact 256th-largest

    // Scale in float domain (IEEE semantics match jnp.where(pred < thresh,.))
    float* const o = out + (size_t)row * ROW_LEN;
#pragma unroll
    for (int c = 0; c < LANE_F4; ++c) {
      const int f4 = c * 32 + lane;
      f32x4 ov;
#pragma unroll
      for (int j = 0; j < 4; ++j) {
        const uint32_t kk = key[c * 4 + j];
        const uint32_t b = (kk & 0x80000000u) ? (kk ^ 0x80000000u) : ~kk;
        const float f = __uint_as_float(b);
        ov[j] = f * ((f < thresh) ? 0.75f : 1.25f);
      }
      store_nt_f4(o + f4 * 4, ov);  // global_store_b128 th:TH_STORE_NT
    }
    parity ^= 1;
  }
}

extern "C" void kernel_launch(void* const* d_in, const int* in_sizes, int n_in,
                              void* d_out, int out_size, void* d_ws,
                              size_t ws_size, hipStream_t stream) {
  (void)n_in;
  (void)out_size;
  (void)d_ws;
  (void)ws_size;
  const float* in = (const float*)d_in[0];
  float* out = (float*)d_out;
  const int n = in_sizes[0];
  const int nrows = n / ROW_LEN;  // 173056 for the reference shape

  int grid = (nrows + WAVES_PER_BLOCK - 1) / WAVES_PER_BLOCK;
  if (grid > 4096) grid = 4096;  // persistent waves -> ~5 rows each, keeps
  if (grid < 1) grid = 1;        // the async double-buffer pipeline busy

  hipLaunchKernelGGL(topk_scale_kernel, dim3(grid), dim3(BLOCK_THREADS), 0,
                     stream, in, out, nrows);
}